// OTHead_40518721470500
// MI455X (gfx1250) — compile-verified
//
#include <hip/hip_runtime.h>
#include <math.h>

// ---------------- problem constants (reference: B=4, N=M=4096, DF=16) -------
#define BATCH 4
#define DF 16

static constexpr float kBlur   = 0.01f;
static constexpr float kInv2   = 1.0f / (2.0f * kBlur * kBlur);   // 5000  = 1/(2 blur^2)
static constexpr float kFscale = 1.0f / (kBlur * kBlur);          // 1e4   = 1/blur^2
static constexpr float kXYs    = 2.0f * kInv2;                    // 1e4   = scale for raw-feature dot

typedef __attribute__((ext_vector_type(2))) float v2f;
typedef __attribute__((ext_vector_type(8))) float v8f;

#ifndef __has_builtin
#define __has_builtin(x) 0
#endif
#if __has_builtin(__builtin_amdgcn_wmma_f32_16x16x4_f32)
#define USE_WMMA_F32X4 1
#else
#define USE_WMMA_F32X4 0
#endif

// ---------------------------------------------------------------------------
// Kernel 1: alpha[b][n] = F/blur^2 - inv2*sum(fea^2), beta likewise for tgt.
// ---------------------------------------------------------------------------
__global__ void k_prep(const float* __restrict__ src_fea,
                       const float* __restrict__ tgt_fea,
                       const float* __restrict__ F_i,
                       const float* __restrict__ G_j,
                       float* __restrict__ alpha,
                       float* __restrict__ beta,
                       int BN, int BM) {
  int idx = blockIdx.x * blockDim.x + threadIdx.x;
  if (idx < BN) {
    const float* f = src_fea + (size_t)idx * DF;
    float s = 0.f;
#pragma unroll
    for (int k = 0; k < DF; ++k) s += f[k] * f[k];
    alpha[idx] = F_i[idx] * kFscale - kInv2 * s;
  } else if (idx < BN + BM) {
    int j = idx - BN;
    const float* f = tgt_fea + (size_t)j * DF;
    float s = 0.f;
#pragma unroll
    for (int k = 0; k < DF; ++k) s += f[k] * f[k];
    beta[j] = G_j[j] * kFscale - kInv2 * s;
  }
}

// ---------------------------------------------------------------------------
// Kernel 2: per (b, 16-row tile): online softmax over all M columns.
// 8 waves/block, each wave owns interleaved 16-col tiles.
// 2*xy term via chained V_WMMA_F32_16X16X4_F32 (K = 16 = 4 chunks of 4).
// Writes 23 Kabsch partial sums per block (deterministic, no atomics).
// ---------------------------------------------------------------------------
__launch_bounds__(256, 2)
__global__ void k_softmax_map(const float* __restrict__ src,
                              const float* __restrict__ tgt,
                              const float* __restrict__ src_fea,
                              const float* __restrict__ tgt_fea,
                              const float* __restrict__ alpha,
                              const float* __restrict__ beta,
                              float* __restrict__ part,   // [B*tiles][24]
                              int N, int M) {
  const int b    = blockIdx.y;
  const int n0   = blockIdx.x * 16;
  const int tid  = threadIdx.x;
  const int wave = tid >> 5;
  const int lane = tid & 31;
  const int lh   = lane & 15;   // column within 16-wide tile / row within half
  const int hi   = lane >> 4;   // which lane-half

  // ---- A fragments (16 rows x DF), constant across the m loop -------------
  // f32 16x4 A layout: lanes 0-15 row=lh {K=4c,4c+1}; lanes 16-31 row=lh {K=4c+2,4c+3}
  v2f afrag[4];
  {
    const float* ap = src_fea + ((size_t)(b * N + n0 + lh)) * DF;
#pragma unroll
    for (int c = 0; c < 4; ++c) {
      int k0 = 4 * c + 2 * hi;
      afrag[c].x = ap[k0];
      afrag[c].y = ap[k0 + 1];
    }
  }
  // alpha per D-row: D VGPR r -> row n0+r (lanes 0-15) / n0+8+r (lanes 16-31)
  float aRow[8];
#pragma unroll
  for (int r = 0; r < 8; ++r) aRow[r] = alpha[b * N + n0 + hi * 8 + r];

  // ---- per-lane online softmax state, one per D VGPR ----------------------
  float mx[8], sm[8], vx[8], vy[8], vz[8];
#pragma unroll
  for (int r = 0; r < 8; ++r) { mx[r] = -3.0e38f; sm[r] = 0.f; vx[r] = 0.f; vy[r] = 0.f; vz[r] = 0.f; }

  for (int m0 = wave * 16; m0 < M; m0 += 8 * 16) {
    const int col  = m0 + lh;                    // this lane's D column
    const float* bp = tgt_fea + ((size_t)(b * M + col)) * DF;
    const float betaV = beta[b * M + col];
    const float* tp = tgt + ((size_t)(b * M + col)) * 3;
    const float tx = tp[0], ty = tp[1], tz = tp[2];

    float sc[8];
#if USE_WMMA_F32X4
    // B (4x16) mirrors A: row K striped across lanes within a VGPR.
    v2f bfrag[4];
#pragma unroll
    for (int c = 0; c < 4; ++c) {
      int k0 = 4 * c + 2 * hi;
      bfrag[c].x = bp[k0];
      bfrag[c].y = bp[k0 + 1];
    }
    v8f dmat = {0.f, 0.f, 0.f, 0.f, 0.f, 0.f, 0.f, 0.f};
#pragma unroll
    for (int c = 0; c < 4; ++c)
      dmat = __builtin_amdgcn_wmma_f32_16x16x4_f32(false, afrag[c], false, bfrag[c],
                                                   (short)0, dmat, false, false);
#pragma unroll
    for (int r = 0; r < 8; ++r) sc[r] = dmat[r];
#else
    // scalar fallback: direct dot per (row, col)
#pragma unroll
    for (int r = 0; r < 8; ++r) {
      const float* arp = src_fea + ((size_t)(b * N + n0 + hi * 8 + r)) * DF;
      float d = 0.f;
#pragma unroll
      for (int k = 0; k < DF; ++k) d += arp[k] * bp[k];
      sc[r] = d;
    }
#endif

#pragma unroll
    for (int r = 0; r < 8; ++r) {
      float s  = aRow[r] + betaV + kXYs * sc[r];
      float nm = fmaxf(mx[r], s);
      float e0 = __expf(mx[r] - nm);   // correction (==1 when max unchanged)
      float p  = __expf(s - nm);
      mx[r] = nm;
      sm[r] = sm[r] * e0 + p;
      vx[r] = vx[r] * e0 + p * tx;
      vy[r] = vy[r] * e0 + p * ty;
      vz[r] = vz[r] * e0 + p * tz;
    }
  }

  // ---- reduce the 16 columns of each lane-half (xor tree, halves disjoint) --
#pragma unroll
  for (int r = 0; r < 8; ++r) {
#pragma unroll
    for (int off = 1; off < 16; off <<= 1) {
      float m2 = __shfl_xor(mx[r], off, 32);
      float s2 = __shfl_xor(sm[r], off, 32);
      float bx = __shfl_xor(vx[r], off, 32);
      float by = __shfl_xor(vy[r], off, 32);
      float bz = __shfl_xor(vz[r], off, 32);
      float nm = fmaxf(mx[r], m2);
      float e1 = __expf(mx[r] - nm), e2 = __expf(m2 - nm);
      mx[r] = nm;
      sm[r] = sm[r] * e1 + s2 * e2;
      vx[r] = vx[r] * e1 + bx * e2;
      vy[r] = vy[r] * e1 + by * e2;
      vz[r] = vz[r] * e1 + bz * e2;
    }
  }

  __shared__ float smem[8][16][5];
  if (lh == 0) {  // lanes 0 and 16 hold the per-half results
#pragma unroll
    for (int r = 0; r < 8; ++r) {
      int row = hi * 8 + r;
      smem[wave][row][0] = mx[r];
      smem[wave][row][1] = sm[r];
      smem[wave][row][2] = vx[r];
      smem[wave][row][3] = vy[r];
      smem[wave][row][4] = vz[r];
    }
  }
  __syncthreads();

  if (wave == 0) {
    float vals[23];
    if (hi == 0) {
      const int row = lh;  // one lane per tile row
      float m = -3.0e38f, s = 0.f, ax = 0.f, ay = 0.f, az = 0.f;
      for (int w = 0; w < 8; ++w) {
        float m2 = smem[w][row][0], s2 = smem[w][row][1];
        float bx = smem[w][row][2], by = smem[w][row][3], bz = smem[w][row][4];
        float nm = fmaxf(m, m2);
        float e1 = __expf(m - nm), e2 = __expf(m2 - nm);
        s  = s * e1 + s2 * e2;
        ax = ax * e1 + bx * e2;
        ay = ay * e1 + by * e2;
        az = az * e1 + bz * e2;
        m  = nm;
      }
      float lse = m + __logf(s);
      float wgt = __expf(lse);               // mapped mass ratio
      float is_ = 1.0f / s;
      float y0 = ax * is_, y1 = ay * is_, y2v = az * is_;   // mapped position
      const float* sp = src + ((size_t)(b * N + n0 + row)) * 3;
      float x0 = sp[0], x1 = sp[1], x2v = sp[2];
      float w2 = wgt * wgt;                  // reference multiplies BOTH hats by w
      vals[0] = wgt;
      vals[1] = wgt * x0;  vals[2] = wgt * x1;  vals[3] = wgt * x2v;
      vals[4] = wgt * y0;  vals[5] = wgt * y1;  vals[6] = wgt * y2v;
      vals[7] = w2;
      vals[8]  = w2 * x0;  vals[9]  = w2 * x1;  vals[10] = w2 * x2v;
      vals[11] = w2 * y0;  vals[12] = w2 * y1;  vals[13] = w2 * y2v;
      float yv[3] = {y0, y1, y2v}, xv[3] = {x0, x1, x2v};
#pragma unroll
      for (int d = 0; d < 3; ++d)
#pragma unroll
        for (int e = 0; e < 3; ++e) vals[14 + 3 * d + e] = w2 * yv[d] * xv[e];
    } else {
#pragma unroll
      for (int k = 0; k < 23; ++k) vals[k] = 0.f;
    }
    // sum the 16 rows (lanes 0-15); lanes 16-31 contribute zeros, halves disjoint
#pragma unroll
    for (int k = 0; k < 23; ++k)
#pragma unroll
      for (int off = 1; off < 16; off <<= 1) vals[k] += __shfl_xor(vals[k], off, 32);

    if (lane == 0) {
      float* p = part + ((size_t)(b * gridDim.x + blockIdx.x)) * 24;
#pragma unroll
      for (int k = 0; k < 23; ++k) p[k] = vals[k];
    }
  }
}

// ---------------------------------------------------------------------------
// Kernel 3: deterministic sum of partials, then rigid solve per batch.
// Reference r = (u*c[:,:,None])@vh  ==  diag(1,1,sign(det A)) * polar(A),
// so compute the orthogonal polar factor via gamma-scaled Newton iteration.
// ---------------------------------------------------------------------------
__global__ void k_finalize(const float* __restrict__ part, int tiles,
                           float* __restrict__ out) {
  const int b = threadIdx.x;
  if (b >= BATCH) return;

  float S[23];
#pragma unroll
  for (int k = 0; k < 23; ++k) S[k] = 0.f;
  for (int t = 0; t < tiles; ++t) {
    const float* p = part + ((size_t)(b * tiles + t)) * 24;
#pragma unroll
    for (int k = 0; k < 23; ++k) S[k] += p[k];
  }

  const float Sw = S[0], Sw2 = S[7];
  float mux[3], muy[3];
#pragma unroll
  for (int d = 0; d < 3; ++d) { mux[d] = S[1 + d] / Sw; muy[d] = S[4 + d] / Sw; }

  // A[d][e] = sum w^2 (y-mu_y)_d (x-mu_x)_e, expanded around uncentered moments
  float A[3][3];
#pragma unroll
  for (int d = 0; d < 3; ++d)
#pragma unroll
    for (int e = 0; e < 3; ++e)
      A[d][e] = S[14 + 3 * d + e] - muy[d] * S[8 + e] - mux[e] * S[11 + d]
              + muy[d] * mux[e] * Sw2;

  const float detA = A[0][0]*(A[1][1]*A[2][2]-A[1][2]*A[2][1])
                   - A[0][1]*(A[1][0]*A[2][2]-A[1][2]*A[2][0])
                   + A[0][2]*(A[1][0]*A[2][1]-A[1][1]*A[2][0]);
  const float sgn = (detA < 0.f) ? -1.f : 1.f;

  // normalize (polar factor invariant to positive scaling)
  float nf = 0.f;
#pragma unroll
  for (int i = 0; i < 3; ++i)
#pragma unroll
    for (int j = 0; j < 3; ++j) nf += A[i][j] * A[i][j];
  nf = sqrtf(nf);
  if (nf < 1e-30f) nf = 1.f;
  float Q[3][3];
#pragma unroll
  for (int i = 0; i < 3; ++i)
#pragma unroll
    for (int j = 0; j < 3; ++j) Q[i][j] = A[i][j] / nf;

  for (int it = 0; it < 16; ++it) {
    float cof[3][3];
    cof[0][0] =  Q[1][1]*Q[2][2] - Q[1][2]*Q[2][1];
    cof[0][1] = -(Q[1][0]*Q[2][2] - Q[1][2]*Q[2][0]);
    cof[0][2] =  Q[1][0]*Q[2][1] - Q[1][1]*Q[2][0];
    cof[1][0] = -(Q[0][1]*Q[2][2] - Q[0][2]*Q[2][1]);
    cof[1][1] =  Q[0][0]*Q[2][2] - Q[0][2]*Q[2][0];
    cof[1][2] = -(Q[0][0]*Q[2][1] - Q[0][1]*Q[2][0]);
    cof[2][0] =  Q[0][1]*Q[1][2] - Q[0][2]*Q[1][1];
    cof[2][1] = -(Q[0][0]*Q[1][2] - Q[0][2]*Q[1][0]);
    cof[2][2] =  Q[0][0]*Q[1][1] - Q[0][1]*Q[1][0];
    float det = Q[0][0]*cof[0][0] + Q[0][1]*cof[0][1] + Q[0][2]*cof[0][2];
    if (fabsf(det) < 1e-25f) break;
    float nq = 0.f, nc = 0.f;
#pragma unroll
    for (int i = 0; i < 3; ++i)
#pragma unroll
      for (int j = 0; j < 3; ++j) { nq += Q[i][j]*Q[i][j]; nc += cof[i][j]*cof[i][j]; }
    // gamma = sqrt(||Q^-1||_F / ||Q||_F) = (nc / (nq*det^2))^(1/4)
    float g = sqrtf(sqrtf(nc / (nq * det * det)));
    float a0 = 0.5f * g;
    float a1 = 0.5f / (g * det);     // (1/g)*Q^{-T} = cof/(g*det)
#pragma unroll
    for (int i = 0; i < 3; ++i)
#pragma unroll
      for (int j = 0; j < 3; ++j) Q[i][j] = a0 * Q[i][j] + a1 * cof[i][j];
  }

  // R = diag(1,1,sgn) * Q ; t = mu_y - R*mu_x
  float R[3][3];
#pragma unroll
  for (int i = 0; i < 3; ++i) {
    float ci = (i == 2) ? sgn : 1.f;
#pragma unroll
    for (int j = 0; j < 3; ++j) R[i][j] = ci * Q[i][j];
  }
#pragma unroll
  for (int i = 0; i < 3; ++i)
#pragma unroll
    for (int j = 0; j < 3; ++j) out[b * 9 + 3 * i + j] = R[i][j];
#pragma unroll
  for (int i = 0; i < 3; ++i) {
    float t = muy[i] - (R[i][0]*mux[0] + R[i][1]*mux[1] + R[i][2]*mux[2]);
    out[BATCH * 9 + b * 3 + i] = t;
  }
}

// ---------------------------------------------------------------------------
extern "C" void kernel_launch(void* const* d_in, const int* in_sizes, int n_in,
                              void* d_out, int out_size, void* d_ws, size_t ws_size,
                              hipStream_t stream) {
  const float* src     = (const float*)d_in[0];
  const float* tgt     = (const float*)d_in[1];
  const float* src_fea = (const float*)d_in[2];
  const float* tgt_fea = (const float*)d_in[3];
  const float* F_i     = (const float*)d_in[4];
  const float* G_j     = (const float*)d_in[5];
  float* out = (float*)d_out;

  const int N = in_sizes[4] / BATCH;   // 4096
  const int M = in_sizes[5] / BATCH;   // 4096
  const int BN = BATCH * N, BM = BATCH * M;
  const int tiles = N / 16;

  float* alpha = (float*)d_ws;                 // [B*N]
  float* beta  = alpha + BN;                   // [B*M]
  float* part  = beta + BM;                    // [B*tiles][24]

  int total = BN + BM;
  k_prep<<<(total + 255) / 256, 256, 0, stream>>>(src_fea, tgt_fea, F_i, G_j,
                                                  alpha, beta, BN, BM);

  dim3 grid(tiles, BATCH);
  k_softmax_map<<<grid, 256, 0, stream>>>(src, tgt, src_fea, tgt_fea,
                                          alpha, beta, part, N, M);

  k_finalize<<<1, 32, 0, stream>>>(part, tiles, out);
}